// GCNNet2_45767171506834
// MI455X (gfx1250) — compile-verified
//
#include <hip/hip_runtime.h>
#include <hip/hip_bf16.h>

// ---------------------------------------------------------------------------
// GCN (4-layer GraphConv + BN + residual + graph-mean readout) for gfx1250.
// Dominant cost is memory (edge gather/scatter ~2.5 GB total @ 23.3 TB/s);
// GEMMs (10.7 GFLOP total) run on V_WMMA_F32_16X16X4_F32 so we keep full f32
// precision at negligible cost since the kernel is bandwidth-bound anyway.
// ---------------------------------------------------------------------------

#define N_NODES  50000
#define N_EDGES  500000
#define N_GRAPHS 100
#define D        146
#define DPAD     160            // 10 tiles of 16; pad cols kept zero
#define NPADE    (N_NODES * DPAD)
#define WPADE    (DPAD * DPAD)  // one padded weight matrix
#define BN_EPS   1e-5f

typedef __attribute__((ext_vector_type(2))) float v2f;
typedef __attribute__((ext_vector_type(8))) float v8f;

__device__ __forceinline__ void atomAddF(float* p, float v) {
  unsafeAtomicAdd(p, v);   // hardware global_atomic_add_f32
}

// ---------------------------------------------------------------------------
// Graph preprocessing
// ---------------------------------------------------------------------------
__global__ void edge_degrees_k(const int* __restrict__ src, const int* __restrict__ dst,
                               float* __restrict__ deg_out, float* __restrict__ deg_in) {
  int e = blockIdx.x * blockDim.x + threadIdx.x;
  if (e < N_EDGES) {
    atomAddF(&deg_out[src[e]], 1.0f);
    atomAddF(&deg_in [dst[e]], 1.0f);
  }
}

__global__ void finalize_norms_k(float* __restrict__ deg_out, float* __restrict__ deg_in) {
  int n = blockIdx.x * blockDim.x + threadIdx.x;
  if (n < N_NODES) {
    deg_out[n] = rsqrtf(fmaxf(deg_out[n], 1.0f));
    deg_in [n] = rsqrtf(fmaxf(deg_in [n], 1.0f));
  }
}

__global__ void node_counts_k(const int* __restrict__ gid, float* __restrict__ cnt) {
  int n = blockIdx.x * blockDim.x + threadIdx.x;
  if (n < N_NODES) atomAddF(&cnt[gid[n]], 1.0f);
}

__global__ void inv_counts_k(float* __restrict__ cnt) {
  int g = threadIdx.x;
  if (g < N_GRAPHS) cnt[g] = 1.0f / fmaxf(cnt[g], 1.0f);
}

// Copy [146x146] weights (W_emb + 4 layer Ws) into zero-padded [160x160].
__global__ void pad_weights_k(const float* __restrict__ W_emb, const float* __restrict__ Ws,
                              float* __restrict__ wpad) {
  int mat = blockIdx.x / DPAD;          // 0..4
  int r   = blockIdx.x % DPAD;          // 0..159
  int c   = threadIdx.x;                // 0..159
  float v = 0.0f;
  if (r < D && c < D) {
    const float* Wsrc = (mat == 0) ? W_emb : (Ws + (size_t)(mat - 1) * D * D);
    v = Wsrc[(size_t)r * D + c];
  }
  wpad[(size_t)mat * WPADE + (size_t)r * DPAD + c] = v;
}

// Copy node features [N x 146] into zero-padded [N x 160].
__global__ void pad_feat_k(const float* __restrict__ x, float* __restrict__ xp) {
  int n = blockIdx.x;
  int c = threadIdx.x;                  // 160 threads
  xp[(size_t)n * DPAD + c] = (c < D) ? x[(size_t)n * D + c] : 0.0f;
}

// ---------------------------------------------------------------------------
// WMMA GEMM: C[N x DPAD] = diag(rowscale) * A[N x DPAD] @ Wp[DPAD x DPAD] (+bias)
// One wave computes one 16x16 output tile; K-loop: 40 steps of K=4 f32 WMMA.
// Fragment layouts per CDNA5 ISA 7.12.2:
//   A 16x4:  lanes 0-15 hold (M=l, K=k,k+1); lanes 16-31 hold (M=l, K=k+2,k+3)
//   B 4x16:  lanes 0-15 hold (K=k..k+1, N=l); lanes 16-31 hold (K=k+2..k+3, N=l)
//   C/D:     VGPR r -> M = r (lanes 0-15) / r+8 (lanes 16-31), N = l
// ---------------------------------------------------------------------------
__global__ void gemm_wmma_k(const float* __restrict__ A, const float* __restrict__ Wp,
                            const float* __restrict__ bias, const float* __restrict__ rowscale,
                            float* __restrict__ C, int nRowTiles, int nColTiles) {
  const int wave   = blockIdx.x * (blockDim.x >> 5) + (threadIdx.x >> 5);
  const int nTiles = nRowTiles * nColTiles;
  if (wave >= nTiles) return;                 // wave-uniform; EXEC stays all-1s
  const int rowTile = wave / nColTiles;
  const int colTile = wave - rowTile * nColTiles;

  const int lane = threadIdx.x & 31;
  const int half = lane >> 4;                 // 0: lanes 0-15, 1: lanes 16-31
  const int l    = lane & 15;
  const int row  = rowTile * 16 + l;          // A row this lane feeds
  const int col0 = colTile * 16;

  const float rs = rowscale ? rowscale[row] : 1.0f;
  const float* __restrict__ arow = A + (size_t)row * DPAD;

  v8f acc = {};
#pragma unroll 8
  for (int k = 0; k < DPAD; k += 4) {
    const int ka = k + 2 * half;
    v2f a, b;
    a.x = arow[ka]     * rs;
    a.y = arow[ka + 1] * rs;
    b.x = Wp[(size_t)ka * DPAD + col0 + l];
    b.y = Wp[(size_t)(ka + 1) * DPAD + col0 + l];
    acc = __builtin_amdgcn_wmma_f32_16x16x4_f32(
        /*neg_a=*/false, a, /*neg_b=*/false, b,
        /*c_mod=*/(short)0, acc, /*reuse_a=*/false, /*reuse_b=*/false);
  }

  const int  ccol = col0 + l;
  const float badd = (bias && ccol < D) ? bias[ccol] : 0.0f;  // pad cols: acc==0, store 0
#pragma unroll
  for (int r = 0; r < 8; ++r) {
    const int orow = rowTile * 16 + r + 8 * half;
    C[(size_t)orow * DPAD + ccol] = acc[r] + badd;
  }
}

// ---------------------------------------------------------------------------
// Edge message passing: agg[dst] += hW[src]; one wave per edge.
// ---------------------------------------------------------------------------
__global__ void scatter_edges_k(const float* __restrict__ hW, const int* __restrict__ src,
                                const int* __restrict__ dst, float* __restrict__ agg) {
  const int wave = blockIdx.x * (blockDim.x >> 5) + (threadIdx.x >> 5);
  if (wave >= N_EDGES) return;
  const int lane = threadIdx.x & 31;
  const size_t soff = (size_t)src[wave] * DPAD;
  const size_t doff = (size_t)dst[wave] * DPAD;
  for (int c = lane; c < D; c += 32)
    atomAddF(&agg[doff + c], hW[soff + c]);
}

// ---------------------------------------------------------------------------
// BatchNorm statistics. t = (agg*norm_dst + b) * snorm_n is recomputed on the
// fly (cheaper than an extra 29MB store+load pass). Thread t owns column t:
// per row, threads read consecutive addresses -> fully coalesced.
// ---------------------------------------------------------------------------
#define STAT_BLOCKS 250
#define STAT_ROWS   (N_NODES / STAT_BLOCKS)   // 200

__global__ void bn_stats_k(const float* __restrict__ agg, const float* __restrict__ norm_dst,
                           const float* __restrict__ bvec, const float* __restrict__ snorm_n,
                           float* __restrict__ colsum, float* __restrict__ colsq) {
  const int c = threadIdx.x;
  if (c >= D) return;
  const float bc = bvec[c];
  float s = 0.0f, sq = 0.0f;
  const int n0 = blockIdx.x * STAT_ROWS;
  for (int r = 0; r < STAT_ROWS; ++r) {
    const int n = n0 + r;
    const float t = (agg[(size_t)n * DPAD + c] * norm_dst[n] + bc) * snorm_n[n];
    s  += t;
    sq += t * t;
  }
  atomAddF(&colsum[c], s);
  atomAddF(&colsq [c], sq);
}

// ---------------------------------------------------------------------------
// Finalize layer: h += relu(BN(t)); fused per-graph mean accumulation into hg.
// ---------------------------------------------------------------------------
#define FIN_ROWS 8

__global__ void finalize_layer_k(const float* __restrict__ agg, float* __restrict__ h,
                                 const float* __restrict__ norm_dst, const float* __restrict__ bvec,
                                 const float* __restrict__ snorm_n,
                                 const float* __restrict__ colsum, const float* __restrict__ colsq,
                                 const float* __restrict__ gamma, const float* __restrict__ beta,
                                 const int* __restrict__ gid, const float* __restrict__ invcnt,
                                 float* __restrict__ hg) {
  const int c = threadIdx.x;
  if (c >= D) return;
  const float invN = 1.0f / (float)N_NODES;
  const float mean = colsum[c] * invN;
  const float var  = fmaxf(colsq[c] * invN - mean * mean, 0.0f);
  const float bnin = rsqrtf(var + BN_EPS) * gamma[c];
  const float bc = bvec[c], bb = beta[c];
  const int n0 = blockIdx.x * FIN_ROWS;
#pragma unroll
  for (int r = 0; r < FIN_ROWS; ++r) {
    const int n = n0 + r;
    const float t = (agg[(size_t)n * DPAD + c] * norm_dst[n] + bc) * snorm_n[n];
    const float v = fmaxf((t - mean) * bnin + bb, 0.0f);
    const float hn = h[(size_t)n * DPAD + c] + v;     // residual
    h[(size_t)n * DPAD + c] = hn;
    const int g = gid[n];
    atomAddF(&hg[(size_t)g * D + c], hn * invcnt[g]); // hg += mean_nodes(h)
  }
}

// ---------------------------------------------------------------------------
// MLP readout: 146 -> 73 -> 36 -> 10 on [100 x 146]; one block per graph.
// ---------------------------------------------------------------------------
__global__ void readout_k(const float* __restrict__ hg,
                          const float* __restrict__ W0, const float* __restrict__ b0,
                          const float* __restrict__ W1, const float* __restrict__ b1,
                          const float* __restrict__ W2, const float* __restrict__ b2,
                          float* __restrict__ out) {
  __shared__ float row[D];
  __shared__ float z1[73];
  __shared__ float z2[36];
  const int g = blockIdx.x;
  const int t = threadIdx.x;
  for (int c = t; c < D; c += blockDim.x) row[c] = hg[(size_t)g * D + c];
  __syncthreads();
  for (int j = t; j < 73; j += blockDim.x) {
    float s = b0[j];
    for (int k = 0; k < D; ++k) s += row[k] * W0[(size_t)k * 73 + j];
    z1[j] = fmaxf(s, 0.0f);
  }
  __syncthreads();
  for (int j = t; j < 36; j += blockDim.x) {
    float s = b1[j];
    for (int k = 0; k < 73; ++k) s += z1[k] * W1[(size_t)k * 36 + j];
    z2[j] = fmaxf(s, 0.0f);
  }
  __syncthreads();
  for (int j = t; j < 10; j += blockDim.x) {
    float s = b2[j];
    for (int k = 0; k < 36; ++k) s += z2[k] * W2[(size_t)k * 10 + j];
    out[(size_t)g * 10 + j] = s;
  }
}

// ---------------------------------------------------------------------------
// Host-side orchestration
// ---------------------------------------------------------------------------
extern "C" void kernel_launch(void* const* d_in, const int* in_sizes, int n_in,
                              void* d_out, int out_size, void* d_ws, size_t ws_size,
                              hipStream_t stream) {
  const float* nodes_feat = (const float*)d_in[0];
  const float* snorm_n    = (const float*)d_in[1];
  const float* W_emb      = (const float*)d_in[2];
  const float* b_emb      = (const float*)d_in[3];
  const float* Ws         = (const float*)d_in[4];
  const float* bs         = (const float*)d_in[5];
  const float* gammas     = (const float*)d_in[6];
  const float* betas      = (const float*)d_in[7];
  const float* W_r0       = (const float*)d_in[8];
  const float* b_r0       = (const float*)d_in[9];
  const float* W_r1       = (const float*)d_in[10];
  const float* b_r1       = (const float*)d_in[11];
  const float* W_r2       = (const float*)d_in[12];
  const float* b_r2       = (const float*)d_in[13];
  const int*   src        = (const int*)d_in[14];
  const int*   dst        = (const int*)d_in[15];
  const int*   gid        = (const int*)d_in[16];
  float*       out        = (float*)d_out;

  // Workspace carve-up (~97 MB of floats).
  float* w      = (float*)d_ws;
  float* bufA   = w;                       // NPADE : padded feats, then agg
  float* bufH   = bufA + NPADE;            // NPADE : current node features h
  float* bufT   = bufH + NPADE;            // NPADE : hW scratch
  float* wpad   = bufT + NPADE;            // 5 * WPADE
  float* nsrc   = wpad + 5 * WPADE;        // N_NODES : norm over out-degree
  float* ndst   = nsrc + N_NODES;          // N_NODES : norm over in-degree
  float* colsum = ndst + N_NODES;          // DPAD
  float* colsq  = colsum + DPAD;           // DPAD
  float* cnt    = colsq + DPAD;            // N_GRAPHS : counts -> inv counts
  float* hg     = cnt + N_GRAPHS;          // N_GRAPHS * D

  // --- init (graph-capture-safe async memsets) ---
  hipMemsetAsync(nsrc, 0, 2ull * N_NODES * sizeof(float), stream);          // nsrc+ndst
  hipMemsetAsync(cnt,  0, (N_GRAPHS + (size_t)N_GRAPHS * D) * sizeof(float), stream); // cnt+hg

  // --- graph preprocessing ---
  edge_degrees_k<<<(N_EDGES + 255) / 256, 256, 0, stream>>>(src, dst, nsrc, ndst);
  node_counts_k<<<(N_NODES + 255) / 256, 256, 0, stream>>>(gid, cnt);
  finalize_norms_k<<<(N_NODES + 255) / 256, 256, 0, stream>>>(nsrc, ndst);
  inv_counts_k<<<1, 128, 0, stream>>>(cnt);

  pad_weights_k<<<5 * DPAD, DPAD, 0, stream>>>(W_emb, Ws, wpad);
  pad_feat_k<<<N_NODES, DPAD, 0, stream>>>(nodes_feat, bufA);

  // --- embedding GEMM: h = X @ W_emb + b_emb ---
  const int nRowTiles = N_NODES / 16;          // 3125
  const int nColTiles = DPAD / 16;             // 10
  const int nTiles    = nRowTiles * nColTiles; // 31250
  const int gemmGrid  = (nTiles + 7) / 8;      // 8 waves / 256-thread block
  gemm_wmma_k<<<gemmGrid, 256, 0, stream>>>(bufA, wpad, b_emb, nullptr, bufH,
                                            nRowTiles, nColTiles);

  // --- 4 GCN layers ---
  for (int i = 0; i < 4; ++i) {
    const float* Wp = wpad + (size_t)(i + 1) * WPADE;
    // hW = (h * norm_src) @ W
    gemm_wmma_k<<<gemmGrid, 256, 0, stream>>>(bufH, Wp, nullptr, nsrc, bufT,
                                              nRowTiles, nColTiles);
    // agg = scatter_add(hW[src] -> dst)
    hipMemsetAsync(bufA, 0, (size_t)NPADE * sizeof(float), stream);
    scatter_edges_k<<<(N_EDGES + 7) / 8, 256, 0, stream>>>(bufT, src, dst, bufA);
    // batch statistics of t = (agg*norm_dst + b) * snorm_n
    hipMemsetAsync(colsum, 0, 2ull * DPAD * sizeof(float), stream);
    bn_stats_k<<<STAT_BLOCKS, DPAD, 0, stream>>>(bufA, ndst, bs + (size_t)i * D,
                                                 snorm_n, colsum, colsq);
    // h += relu(BN(t));  hg += mean_nodes(h)
    finalize_layer_k<<<N_NODES / FIN_ROWS, DPAD, 0, stream>>>(
        bufA, bufH, ndst, bs + (size_t)i * D, snorm_n, colsum, colsq,
        gammas + (size_t)i * D, betas + (size_t)i * D, gid, cnt, hg);
  }

  // --- MLP readout ---
  readout_k<<<N_GRAPHS, 128, 0, stream>>>(hg, W_r0, b_r0, W_r1, b_r1, W_r2, b_r2, out);
}